// SumAggregator_43327630082574
// MI455X (gfx1250) — compile-verified
//
#include <hip/hip_runtime.h>
#include <stdint.h>

// ---------------------------------------------------------------------------
// GNN layer: out[dst] = sum_edges h[src],  h = x @ W^T + b
//   x: [50000,128] f32, W: [128,128] f32 (row-major [out][in]), b: [128] f32
//   edge_index: int64 [2, 1.6M]  (row 0 = dst, row 1 = src)
// Strategy (MI455X):
//   * h (25.6MB) and out (25.6MB) are L2-resident (192MB L2) -> scatter phase
//     is L2 gather + f32 atomic bound, not HBM bound. One wave32 per edge:
//     coalesced float4 gather of h[src], 4 atomicAdd per lane into out[dst].
//   * Projection uses V_WMMA_F32_16X16X4_F32 (exact f32 path; GEMM is only
//     1.6 GFLOP so precision, not matrix throughput, is what matters).
// ---------------------------------------------------------------------------

typedef __attribute__((ext_vector_type(2))) float v2f;
typedef __attribute__((ext_vector_type(8))) float v8f;

#define IN_CH  128
#define OUT_CH 128

// ---------------------------------------------------------------------------
// Zero-init of the output accumulator (float4 stores).
// ---------------------------------------------------------------------------
__global__ void zero_f32_kernel(float* __restrict__ p, int n4) {
    int i = blockIdx.x * blockDim.x + threadIdx.x;
    if (i < n4) {
        ((float4*)p)[i] = make_float4(0.f, 0.f, 0.f, 0.f);
    }
}

// ---------------------------------------------------------------------------
// h = x @ W^T + b via V_WMMA_F32_16X16X4_F32 (wave32; one 16x16 tile / wave).
// Block = 256 threads = 8 waves -> one block covers 16 rows x 128 cols.
//
// A tile (16x4 f32, 2 VGPRs/lane): lane L<16 supplies row M=L, K={k,k+1};
//   lane L>=16 supplies row M=L-16, K={k+2,k+3}.
// B tile (4x16 f32, symmetric): lane supplies col N=L%16, same K split.
//   B[k][n] = W[n][k] (W row-major [out][in] == column-major B).
// C/D (8 VGPRs): row = v + (L>=16 ? 8 : 0), col = L%16.
// ---------------------------------------------------------------------------
__global__ void __launch_bounds__(256)
gemm_bias_wmma_kernel(const float* __restrict__ x,
                      const float* __restrict__ W,
                      const float* __restrict__ bias,
                      float* __restrict__ h,
                      int n_nodes) {
    const int lane    = threadIdx.x & 31;
    const int wave    = threadIdx.x >> 5;          // 0..7 -> column tile
    const int rowTile = blockIdx.x * 16;
    const int colTile = wave * 16;

    const int mr = lane & 15;                      // row (A) / col (B) in tile
    const int kb = (lane >> 4) << 1;               // 0 or 2: K sub-offset

    // Clamp for safety if n_nodes not multiple of 16 (WMMA needs EXEC all-1s,
    // so keep every lane live and clamp the load address instead).
    int arow = rowTile + mr;
    if (arow >= n_nodes) arow = n_nodes - 1;

    const float* __restrict__ xrow = x + (size_t)arow * IN_CH;
    const float* __restrict__ wrow = W + (size_t)(colTile + mr) * IN_CH;

    v8f c = {};
    #pragma unroll
    for (int k = 0; k < IN_CH; k += 4) {
        v2f a, b;
        a.x = xrow[k + kb];
        a.y = xrow[k + kb + 1];
        b.x = wrow[k + kb];
        b.y = wrow[k + kb + 1];
        // D = A(16x4) * B(4x16) + C, all f32
        c = __builtin_amdgcn_wmma_f32_16x16x4_f32(
                /*neg_a=*/false, a, /*neg_b=*/false, b,
                /*c_mod=*/(short)0, c, /*reuse_a=*/false, /*reuse_b=*/false);
    }

    const int n     = colTile + (lane & 15);
    const int mbase = rowTile + ((lane >> 4) << 3);
    const float bv  = bias[n];
    float* __restrict__ hp = h + (size_t)mbase * OUT_CH + n;

    // Uniform guard: only the final (partial) row tile needs per-row checks.
    // With n_nodes % 16 == 0 this is always the fast path.
    if (rowTile + 16 <= n_nodes) {
        #pragma unroll
        for (int v = 0; v < 8; ++v) {
            hp[(size_t)v * OUT_CH] = c[v] + bv;
        }
    } else {
        #pragma unroll
        for (int v = 0; v < 8; ++v) {
            if (mbase + v < n_nodes) {
                hp[(size_t)v * OUT_CH] = c[v] + bv;
            }
        }
    }
}

// ---------------------------------------------------------------------------
// Edge scatter: one wave32 per edge. Each lane moves 4 channels (float4
// gather from h[src], 4x global_atomic_add_f32 into out[dst]).
// 128 ch / 32 lanes = float4 per lane -> 512B coalesced per wave.
// ---------------------------------------------------------------------------
__global__ void __launch_bounds__(256)
scatter_add_kernel(const int64_t* __restrict__ edge_index,
                   const float* __restrict__ h,
                   float* __restrict__ out,
                   int n_edges) {
    const int lane = threadIdx.x & 31;
    const int wave = threadIdx.x >> 5;
    const int e    = blockIdx.x * 8 + wave;        // 8 waves per block
    if (e >= n_edges) return;

    const int64_t dst = edge_index[e];             // row 0 = destination
    const int64_t src = edge_index[(size_t)n_edges + e]; // row 1 = source

    const float4 v = ((const float4*)(h + (size_t)src * OUT_CH))[lane];
    float* __restrict__ od = out + (size_t)dst * OUT_CH + lane * 4;

    atomicAdd(od + 0, v.x);
    atomicAdd(od + 1, v.y);
    atomicAdd(od + 2, v.z);
    atomicAdd(od + 3, v.w);
}

// ---------------------------------------------------------------------------
extern "C" void kernel_launch(void* const* d_in, const int* in_sizes, int n_in,
                              void* d_out, int out_size, void* d_ws, size_t ws_size,
                              hipStream_t stream) {
    const float*   x  = (const float*)d_in[0];
    const int64_t* ei = (const int64_t*)d_in[1];   // int64 per reference
    const float*   W  = (const float*)d_in[2];
    const float*   b  = (const float*)d_in[3];
    float*         out = (float*)d_out;
    float*         h   = (float*)d_ws;             // needs 50000*128*4 = 25.6MB

    const int n_nodes = in_sizes[0] / IN_CH;       // 50000
    const int n_edges = in_sizes[1] / 2;           // 1600000

    // 1) out = 0
    {
        const int n4 = out_size / 4;
        zero_f32_kernel<<<(n4 + 255) / 256, 256, 0, stream>>>(out, n4);
    }

    // 2) h = x @ W^T + b   (one 16-row stripe per block, 8 col-tile waves)
    {
        const int nblocks = (n_nodes + 15) / 16;   // 3125
        gemm_bias_wmma_kernel<<<nblocks, 256, 0, stream>>>(x, W, b, h, n_nodes);
    }

    // 3) out[dst] += h[src]  (one wave per edge)
    {
        const int nblocks = (n_edges + 7) / 8;     // 200000
        scatter_add_kernel<<<nblocks, 256, 0, stream>>>(ei, h, out, n_edges);
    }
}